// CustomLoss_1915555414694
// MI455X (gfx1250) — compile-verified
//
#include <hip/hip_runtime.h>

#define T_LEN     4096
#define NTHREADS  256
#define CHUNK     (T_LEN / NTHREADS)   // 16
#define MIN_GAP   5
#define MIN_SIZE  5

typedef __attribute__((ext_vector_type(2))) float v2f;
typedef __attribute__((ext_vector_type(8))) float v8f;

__device__ __forceinline__ float relu0(float v) { return v > 0.0f ? v : 0.0f; }

// One block per batch. Separable max: best = max_i ( a[i] + suffmax_{j>=i+5} b[j] ).
__global__ void __launch_bounds__(NTHREADS)
custom_loss_batch(const float* __restrict__ w_phi,
                  const float* __restrict__ y,
                  const int*   __restrict__ eps_p,
                  float*       __restrict__ partial)
{
    __shared__ float xs[T_LEN];      // staged x
    __shared__ float sfx[T_LEN];     // suffix max of b[]
    __shared__ float cm[NTHREADS];   // per-chunk max of b[]
    __shared__ float red[NTHREADS];  // block max-reduction

    const int   b    = blockIdx.x;
    const int   tid  = threadIdx.x;
    const float epsf = (float)eps_p[0];
    const float y0   = y[2 * b + 0];
    const float y1   = y[2 * b + 1];
    const float* xg  = w_phi + (size_t)b * T_LEN;

    // ---- Stage x[b, :] into LDS via CDNA5 async global->LDS DMA ----
#if defined(__gfx1250__)
    for (int r = 0; r < (T_LEN / (NTHREADS * 4)); ++r) {        // 4 x b128 per thread
        const int e = (r * NTHREADS + tid) * 4;                 // float index, 16B aligned
        const float* gp = xg + e;
        // Flat pointer to __shared__ is {SHARED_BASE_hi, lds_offset}; low 32 bits
        // are exactly the LDS byte offset the instruction's VDST operand wants.
        unsigned lds_off = (unsigned)(unsigned long long)(uintptr_t)(xs + e);
        asm volatile("global_load_async_to_lds_b128 %0, %1, off"
                     :: "v"(lds_off), "v"(gp)
                     : "memory");
    }
#if __has_builtin(__builtin_amdgcn_s_wait_asynccnt)
    __builtin_amdgcn_s_wait_asynccnt(0);
#else
    asm volatile("s_wait_asynccnt 0" ::: "memory");
#endif
#else
    for (int r = 0; r < (T_LEN / (NTHREADS * 4)); ++r) {
        const int e = (r * NTHREADS + tid) * 4;
        *(float4*)(xs + e) = *(const float4*)(xg + e);
    }
#endif
    __syncthreads();

    // ---- Pass A: per-chunk max of b[j] = x[j] + 0.5*relu(|y1-j|-eps) ----
    const int i0 = tid * CHUNK;
    float m = -3.402823466e38f;
    #pragma unroll
    for (int k = 0; k < CHUNK; ++k) {
        const int i = i0 + k;
        const float bb = xs[i] + 0.5f * relu0(fabsf(y1 - (float)i) - epsf);
        m = fmaxf(m, bb);
    }
    cm[tid] = m;
    __syncthreads();

    // ---- Pass B: exclusive suffix max over chunk maxima ----
    float suff = -3.402823466e38f;
    for (int t2 = tid + 1; t2 < NTHREADS; ++t2) suff = fmaxf(suff, cm[t2]);

    // ---- Pass C: fill suffix-max array for own chunk (back-to-front) ----
    float run = suff;
    #pragma unroll
    for (int k = CHUNK - 1; k >= 0; --k) {
        const int i = i0 + k;
        const float bb = xs[i] + 0.5f * relu0(fabsf(y1 - (float)i) - epsf);
        run = fmaxf(run, bb);
        sfx[i] = run;
    }
    __syncthreads();

    // ---- Pass D: best = max_i ( a[i] + sfx[i+MIN_SIZE] ), i in [MIN_GAP, T-1-MIN_SIZE] ----
    float best = -3.402823466e38f;
    #pragma unroll
    for (int k = 0; k < CHUNK; ++k) {
        const int i = i0 + k;
        if (i >= MIN_GAP && (i + MIN_SIZE) < T_LEN) {
            const float a = xs[i] + 0.5f * relu0(fabsf(y0 - (float)i) - epsf);
            best = fmaxf(best, a + sfx[i + MIN_SIZE]);
        }
    }
    if (tid == 0) {
        // init = a[1] + b[1+MIN_SIZE]
        const float a1 = xs[1] + 0.5f * relu0(fabsf(y0 - 1.0f) - epsf);
        const float b6 = xs[1 + MIN_SIZE] +
                         0.5f * relu0(fabsf(y1 - (float)(1 + MIN_SIZE)) - epsf);
        best = fmaxf(best, a1 + b6);
    }

    // ---- Block max reduction ----
    red[tid] = best;
    __syncthreads();
    for (int s = NTHREADS / 2; s > 0; s >>= 1) {
        if (tid < s) red[tid] = fmaxf(red[tid], red[tid + s]);
        __syncthreads();
    }
    if (tid == 0) partial[b] = red[0];
}

// Mean of 8 partials -> d_out[0]. Exact f32 sum done on the matrix core:
// D = A x B + 0 with A rows 0..1 holding the 8 partials and B = all-ones,
// so D[0][0] + D[1][0] = sum(partials). One full wave32, EXEC all ones.
__global__ void __launch_bounds__(32)
finalize_mean(const float* __restrict__ partial, float* __restrict__ out)
{
#if defined(__gfx1250__) && __has_builtin(__builtin_amdgcn_wmma_f32_16x16x4_f32)
    const int lane = threadIdx.x;            // 0..31
    const int mrow = lane & 15;              // M index this lane carries for A
    const int k0   = (lane >> 4) ? 2 : 0;    // VGPR0 holds K=0 (lanes 0-15) / K=2 (16-31)

    v2f a; a.x = 0.0f; a.y = 0.0f;
    if (mrow < 2) {                          // A[0][0..3]=p0..p3, A[1][0..3]=p4..p7
        a.x = partial[mrow * 4 + k0 + 0];
        a.y = partial[mrow * 4 + k0 + 1];
    }
    v2f bones; bones.x = 1.0f; bones.y = 1.0f;   // B == 1 everywhere (layout-independent)
    v8f c = {};
    c = __builtin_amdgcn_wmma_f32_16x16x4_f32(
            /*neg_a=*/false, a, /*neg_b=*/false, bones,
            /*c_mod=*/(short)0, c, /*reuse_a=*/false, /*reuse_b=*/false);
    // Lane 0: c[0] = D[M=0][N=0], c[1] = D[M=1][N=0]
    if (lane == 0) out[0] = (c[0] + c[1]) * 0.125f;
#else
    if (threadIdx.x == 0) {
        float s = 0.0f;
        for (int i = 0; i < 8; ++i) s += partial[i];
        out[0] = s * 0.125f;
    }
#endif
}

extern "C" void kernel_launch(void* const* d_in, const int* in_sizes, int n_in,
                              void* d_out, int out_size, void* d_ws, size_t ws_size,
                              hipStream_t stream)
{
    const float* w_phi = (const float*)d_in[0];   // (8,1,4096) f32
    const float* y     = (const float*)d_in[1];   // (8,2) f32
    const int*   eps   = (const int*)d_in[2];     // scalar int
    float*       out   = (float*)d_out;           // 1 f32
    float*       part  = (float*)d_ws;            // 8 f32 scratch

    custom_loss_batch<<<dim3(8), dim3(NTHREADS), 0, stream>>>(w_phi, y, eps, part);
    finalize_mean<<<dim3(1), dim3(32), 0, stream>>>(part, out);
}